// SMPL_24206435680942
// MI455X (gfx1250) — compile-verified
//
#include <hip/hip_runtime.h>
#include <math.h>

#define SMPL_B  512
#define SMPL_V  6890
#define SMPL_J  24
#define SMPL_NB 10
#define SMPL_NP 207
#define KTOT    220      // 10 + 207 + 3 zero pad
#define KK2     110      // KTOT/2
#define NSTEP   55       // KTOT/4
#define PROWS   48       // panel rows = 16 vertices * 3 dims
#define PSTR    210      // posedirs panel LDS row stride (207 + 3 TDM zero pad)
#define SSTR    12       // shapedirs panel LDS row stride (10 + 2 TDM zero pad)

// workspace layout (float offsets)
#define WS_ALIN 0                      // [B][220]   A rows (betas|posemap|0)
#define WS_A2   112640                 // [110][B]   float2 k-pairs, K-major
#define WS_ROTS 225280                 // [B][24][9]
#define WS_JS   335872                 // [24][3][11]  (slot 10 = v_template part)
#define WS_GT2  336664                 // [12][B*12] float2 j-pairs of G

typedef float v2f __attribute__((ext_vector_type(2)));
typedef float v4f __attribute__((ext_vector_type(4)));
typedef float v8f __attribute__((ext_vector_type(8)));
typedef unsigned int v4u __attribute__((ext_vector_type(4)));
typedef int v4i __attribute__((ext_vector_type(4)));
typedef int v8i __attribute__((ext_vector_type(8)));

#ifndef __has_builtin
#define __has_builtin(x) 0
#endif
#if __has_builtin(__builtin_amdgcn_tensor_load_to_lds)
#define HAVE_TDM 1
#else
#define HAVE_TDM 0
#endif
#if __has_include(<hip/amd_detail/amd_gfx1250_TDM.h>)
#define TDM_6ARG 1
#else
#define TDM_6ARG 0
#endif

__device__ __forceinline__ v8f wmma_f32(v2f a, v2f b, v8f c) {
  // D = A(16x4 f32) * B(4x16 f32) + C(16x16 f32)
  return __builtin_amdgcn_wmma_f32_16x16x4_f32(false, a, false, b, (short)0, c,
                                               false, false);
}

#if HAVE_TDM
// 2D tile load Global->LDS via Tensor Data Mover.
// tile_d0 may exceed tensor_d0 and tile rows may exceed tensor_d1: TDM
// zero-fills out-of-tensor elements, giving padded LDS rows + ragged-edge
// zeroing for free.
__device__ __forceinline__ void tdm_load_2d(unsigned lds_addr,
                                            unsigned long long gaddr,
                                            unsigned tensor_d0, unsigned tensor_d1,
                                            unsigned tile_d0, unsigned tile_d1,
                                            unsigned stride0) {
  v4u g0;
  g0[0] = 1u;                                   // count=1, user mode
  g0[1] = lds_addr;                             // LDS byte address
  g0[2] = (unsigned)(gaddr & 0xffffffffull);    // global addr lo
  g0[3] = (unsigned)((gaddr >> 32) & 0x1ffffffull) | (2u << 30);  // hi | type=2
  v8i g1;
  g1[0] = (int)(2u << 16);                      // wg_mask=0, data_size=4B
  g1[1] = (int)((tensor_d0 & 0xffffu) << 16);   // tensor_dim0[15:0]
  g1[2] = (int)((tensor_d0 >> 16) | ((tensor_d1 & 0xffffu) << 16));
  g1[3] = (int)((tensor_d1 >> 16) | (tile_d0 << 16));
  g1[4] = (int)(tile_d1 & 0xffffu);             // tile_dim1 (tile_dim2=0)
  g1[5] = (int)stride0;                         // tensor_dim0_stride lo32
  g1[6] = 0;
  g1[7] = 0;
  v4i g2; g2[0] = 1; g2[1] = 0; g2[2] = 0; g2[3] = 0;   // tensor_dim2=1
  v4i g3; g3[0] = 0; g3[1] = 0; g3[2] = 0; g3[3] = 0;
#if TDM_6ARG
  v8i gz; gz[0]=0; gz[1]=0; gz[2]=0; gz[3]=0; gz[4]=0; gz[5]=0; gz[6]=0; gz[7]=0;
  __builtin_amdgcn_tensor_load_to_lds(g0, g1, g2, g3, gz, 0);
#else
  __builtin_amdgcn_tensor_load_to_lds(g0, g1, g2, g3, 0);
#endif
}
#endif

// ---------------------------------------------------------------- kernel 0
__global__ void k_rots(const float* __restrict__ body_pose,
                       const float* __restrict__ betas,
                       float* __restrict__ ws) {
  int t = blockIdx.x * 256 + threadIdx.x;
  if (t >= SMPL_B * SMPL_J) return;
  int b = t / SMPL_J, jj = t - b * SMPL_J;
  const float* ap = body_pose + b * 72 + jj * 3;
  float x = ap[0], y = ap[1], z = ap[2];
  float ax = x + 1e-8f, ay = y + 1e-8f, az = z + 1e-8f;
  float angle = sqrtf(ax * ax + ay * ay + az * az);
  float inv = 1.0f / angle;
  float h = 0.5f * angle;
  float sw = sinf(h);
  float qw = cosf(h), qx = sw * x * inv, qy = sw * y * inv, qz = sw * z * inv;
  float qn = rsqrtf(qw * qw + qx * qx + qy * qy + qz * qz);
  qw *= qn; qx *= qn; qy *= qn; qz *= qn;
  float w2 = qw * qw, x2 = qx * qx, y2 = qy * qy, z2 = qz * qz;
  float wx = qw * qx, wy = qw * qy, wz = qw * qz;
  float xy = qx * qy, xz = qx * qz, yz = qy * qz;
  float r[9];
  r[0] = w2 + x2 - y2 - z2; r[1] = 2 * xy - 2 * wz;     r[2] = 2 * wy + 2 * xz;
  r[3] = 2 * wz + 2 * xy;   r[4] = w2 - x2 + y2 - z2;   r[5] = 2 * yz - 2 * wx;
  r[6] = 2 * xz - 2 * wy;   r[7] = 2 * wx + 2 * yz;     r[8] = w2 - x2 - y2 + z2;
  float* rot = ws + WS_ROTS + t * 9;
#pragma unroll
  for (int e = 0; e < 9; ++e) rot[e] = r[e];
  float* A = ws + WS_ALIN + b * KTOT;
  if (jj == 0) {
#pragma unroll
    for (int s = 0; s < SMPL_NB; ++s) A[s] = betas[b * SMPL_NB + s];
    A[217] = 0.f; A[218] = 0.f; A[219] = 0.f;
  } else {
    float* Aj = A + 10 + (jj - 1) * 9;
#pragma unroll
    for (int e = 0; e < 9; ++e)
      Aj[e] = r[e] - ((e == 0 || e == 4 || e == 8) ? 1.0f : 0.0f);
  }
}

// ---------------------------------------------------------------- kernel 1
__global__ void k_transA(float* __restrict__ ws) {
  int e = blockIdx.x * 256 + threadIdx.x;   // e = kk2*512 + b, 110*512 total
  int kk2 = e >> 9, b = e & 511;
  v2f r;
  r.x = ws[WS_ALIN + b * KTOT + 2 * kk2];
  r.y = ws[WS_ALIN + b * KTOT + 2 * kk2 + 1];
  ((v2f*)(ws + WS_A2))[e] = r;
}

// ---------------------------------------------------------------- kernel 2
// one wave per (joint, dim, shape-slot); coalesced strided partials + shfl.
__global__ void k_js(const float* __restrict__ Jreg,
                     const float* __restrict__ sdirs,
                     const float* __restrict__ vtpl,
                     float* __restrict__ ws) {
  int wid = (blockIdx.x * 256 + threadIdx.x) >> 5;
  int lane = threadIdx.x & 31;
  if (wid >= SMPL_J * 3 * 11) return;
  int jj = wid / 33, rr = wid - jj * 33, d = rr / 11, s = rr - d * 11;
  const float* Jr = Jreg + jj * SMPL_V;
  float acc = 0.f;
  if (s < 10) {
    const float* sp = sdirs + d * 10 + s;
    for (int v = lane; v < SMPL_V; v += 32) acc += Jr[v] * sp[v * 30];
  } else {
    const float* vp = vtpl + d;
    for (int v = lane; v < SMPL_V; v += 32) acc += Jr[v] * vp[v * 3];
  }
#pragma unroll
  for (int m = 16; m; m >>= 1) acc += __shfl_xor(acc, m, 32);
  if (lane == 0) ws[WS_JS + wid] = acc;
}

// ---------------------------------------------------------------- kernel 3
__global__ void k_chain(const float* __restrict__ betas,
                        const float* __restrict__ trans,
                        float* __restrict__ ws,
                        float* __restrict__ out) {
  __shared__ float res[64][289];      // per-thread 24 x (3x4) transforms
  int b = blockIdx.x * 64 + threadIdx.x;
  if (b >= SMPL_B) return;
  float be[10];
#pragma unroll
  for (int s = 0; s < 10; ++s) be[s] = betas[b * 10 + s];
  float jx[24], jy[24], jz[24];
#pragma unroll
  for (int jj = 0; jj < 24; ++jj) {
    float vv[3];
#pragma unroll
    for (int d = 0; d < 3; ++d) {
      const float* Jp = ws + WS_JS + (jj * 3 + d) * 11;
      float a = Jp[10];
#pragma unroll
      for (int s = 0; s < 10; ++s) a += Jp[s] * be[s];
      vv[d] = a;
    }
    jx[jj] = vv[0]; jy[jj] = vv[1]; jz[jj] = vv[2];
  }
  float* R = &res[threadIdx.x][0];
  {
    const float* r0 = ws + WS_ROTS + (b * 24 + 0) * 9;
#pragma unroll
    for (int m = 0; m < 3; ++m) {
      R[m * 4 + 0] = r0[m * 3 + 0];
      R[m * 4 + 1] = r0[m * 3 + 1];
      R[m * 4 + 2] = r0[m * 3 + 2];
    }
    R[3] = jx[0]; R[7] = jy[0]; R[11] = jz[0];
  }
  const int PAR[24] = {-1, 0, 0, 0, 1, 2, 3, 4, 5, 6, 7, 8,
                       9, 9, 9, 12, 13, 14, 16, 17, 18, 19, 20, 21};
#pragma unroll
  for (int i = 1; i < 24; ++i) {
    int p = PAR[i];
    const float* ri = ws + WS_ROTS + (b * 24 + i) * 9;
    float* rp = &res[threadIdx.x][p * 12];
    float* ro = &res[threadIdx.x][i * 12];
    float dx = jx[i] - jx[p], dy = jy[i] - jy[p], dz = jz[i] - jz[p];
#pragma unroll
    for (int m = 0; m < 3; ++m) {
      float p0 = rp[m * 4 + 0], p1 = rp[m * 4 + 1], p2 = rp[m * 4 + 2],
            p3 = rp[m * 4 + 3];
      ro[m * 4 + 0] = p0 * ri[0] + p1 * ri[3] + p2 * ri[6];
      ro[m * 4 + 1] = p0 * ri[1] + p1 * ri[4] + p2 * ri[7];
      ro[m * 4 + 2] = p0 * ri[2] + p1 * ri[5] + p2 * ri[8];
      ro[m * 4 + 3] = p0 * dx + p1 * dy + p2 * dz + p3;
    }
  }
  float t0 = trans[b * 3 + 0], t1 = trans[b * 3 + 1], t2 = trans[b * 3 + 2];
  float* jout = out + (size_t)SMPL_B * SMPL_V * 3;
  float* G2 = ws + WS_GT2;
#pragma unroll
  for (int i = 0; i < 24; ++i) {
    float* r = &res[threadIdx.x][i * 12];
    jout[b * 72 + i * 3 + 0] = r[3] + t0;
    jout[b * 72 + i * 3 + 1] = r[7] + t1;
    jout[b * 72 + i * 3 + 2] = r[11] + t2;
    float jxi = jx[i], jyi = jy[i], jzi = jz[i];
#pragma unroll
    for (int m = 0; m < 3; ++m) {
      float tp = r[m * 4 + 3] -
                 (r[m * 4 + 0] * jxi + r[m * 4 + 1] * jyi + r[m * 4 + 2] * jzi);
#pragma unroll
      for (int cc = 0; cc < 4; ++cc) {
        float val = (cc < 3) ? r[m * 4 + cc] : tp;
        int col = m * 4 + cc;
        G2[(((i >> 1) * (SMPL_B * 12)) + b * 12 + col) * 2 + (i & 1)] = val;
      }
    }
  }
}

// ---------------------------------------------------------------- kernel 4
__global__ void __launch_bounds__(256) k_main(
    const float* __restrict__ ws, const float* __restrict__ sdirs,
    const float* __restrict__ pdirs, const float* __restrict__ vtpl,
    const float* __restrict__ wgt, const float* __restrict__ trans,
    float* __restrict__ out) {
  __shared__ __align__(16) float panS[PROWS * SSTR];   // 2.3 KB shapedirs rows
  __shared__ __align__(16) float panP[PROWS * PSTR];   // 40.3 KB posedirs rows
  __shared__ float vtl[48];
  __shared__ float Pst[8][768];                        // per-wave v_posed tile
  __shared__ __align__(16) float Tst[8][1536];         // per-wave T tile

  const int v0 = blockIdx.x * 16;
  const int tid = threadIdx.x;
  const int rows_valid = (SMPL_V - v0) * 3 < PROWS ? (SMPL_V - v0) * 3 : PROWS;

  // warm L2 for the skinning-matrix panel (read by every block)
  __builtin_prefetch((const char*)(ws + WS_GT2) + tid * 2304, 0, 0);

#if HAVE_TDM
  if (tid < 32) {  // wave 0 drives the Tensor Data Mover
    tdm_load_2d((unsigned)(uintptr_t)&panS[0],
                (unsigned long long)(uintptr_t)sdirs + (unsigned long long)v0 * 120ull,
                10u, (unsigned)rows_valid, SSTR, PROWS, 10u);
    tdm_load_2d((unsigned)(uintptr_t)&panP[0],
                (unsigned long long)(uintptr_t)pdirs + (unsigned long long)v0 * 2484ull,
                207u, (unsigned)rows_valid, PSTR, PROWS, 207u);
    __builtin_amdgcn_s_wait_tensorcnt(0);
  }
#else
  for (int e = tid; e < PROWS * SSTR; e += 256) {
    int n = e / SSTR, c = e - n * SSTR;
    panS[e] = (n < rows_valid && c < 10)
                  ? sdirs[(size_t)(v0 * 3 + n) * 10 + c] : 0.f;
  }
  for (int e = tid; e < PROWS * PSTR; e += 256) {
    int n = e / PSTR, c = e - n * PSTR;
    panP[e] = (n < rows_valid && c < 207)
                  ? pdirs[(size_t)(v0 * 3 + n) * 207 + c] : 0.f;
  }
#endif
  for (int n = tid; n < 48; n += 256) {
    int v = v0 + n / 3;
    vtl[n] = (v < SMPL_V) ? vtpl[v * 3 + (n % 3)] : 0.f;
  }
  __syncthreads();

  const int lane = tid & 31;
  const int w = tid >> 5;
  const int hi = lane >> 4;
  const int nl = lane & 15;

  const v2f* A2 = (const v2f*)(ws + WS_A2);
  const v2f* G2 = (const v2f*)(ws + WS_GT2);

  // per-lane panel row bases (include the 2*hi k-sublane offset)
  const float* bS0 = panS + nl * SSTR + 2 * hi;
  const float* bS1 = panS + (16 + nl) * SSTR + 2 * hi;
  const float* bS2 = panS + (32 + nl) * SSTR + 2 * hi;
  const float* bP0 = panP + nl * PSTR + 2 * hi;
  const float* bP1 = panP + (16 + nl) * PSTR + 2 * hi;
  const float* bP2 = panP + (32 + nl) * PSTR + 2 * hi;

  // weights A-frags: constant across all b-tiles of this block
  v2f awf[6];
  {
    int vv = v0 + nl; if (vv >= SMPL_V) vv = SMPL_V - 1;
#pragma unroll
    for (int ks = 0; ks < 6; ++ks) {
      int k = 4 * ks + 2 * hi;
      awf[ks] = *(const v2f*)(wgt + vv * 24 + k);
    }
  }

  for (int t = 0; t < 4; ++t) {
    const int b0 = (w + 8 * t) * 16;

    // ---- GEMM1: v_posed displacement = [betas|posemap] x [shapedirs|posedirs]
    v8f acc[3];
#pragma unroll
    for (int s = 0; s < 3; ++s)
#pragma unroll
      for (int r = 0; r < 8; ++r) acc[s][r] = 0.f;
    // ks 0..1: pure shapedirs panel (k = 0..7)
#pragma unroll
    for (int ks = 0; ks < 2; ++ks) {
      v2f ag = A2[(2 * ks + hi) * SMPL_B + b0 + nl];
      acc[0] = wmma_f32(ag, *(const v2f*)(bS0 + 4 * ks), acc[0]);
      acc[1] = wmma_f32(ag, *(const v2f*)(bS1 + 4 * ks), acc[1]);
      acc[2] = wmma_f32(ag, *(const v2f*)(bS2 + 4 * ks), acc[2]);
    }
    // ks 2: straddles shapedirs (k=8, hi=0) and posedirs (k=10, hi=1)
    {
      v2f ag = A2[(4 + hi) * SMPL_B + b0 + nl];
      const float* q0 = hi ? (panP + nl * PSTR) : (panS + nl * SSTR + 8);
      const float* q1 = hi ? (panP + (16 + nl) * PSTR) : (panS + (16 + nl) * SSTR + 8);
      const float* q2 = hi ? (panP + (32 + nl) * PSTR) : (panS + (32 + nl) * SSTR + 8);
      acc[0] = wmma_f32(ag, *(const v2f*)q0, acc[0]);
      acc[1] = wmma_f32(ag, *(const v2f*)q1, acc[1]);
      acc[2] = wmma_f32(ag, *(const v2f*)q2, acc[2]);
    }
    // ks 3..54: pure posedirs panel (k = 12..219; cols >=207 are TDM zeros)
    for (int ks = 3; ks < NSTEP; ++ks) {
      v2f ag = A2[(2 * ks + hi) * SMPL_B + b0 + nl];
      int o = 4 * ks - 10;
      acc[0] = wmma_f32(ag, *(const v2f*)(bP0 + o), acc[0]);
      acc[1] = wmma_f32(ag, *(const v2f*)(bP1 + o), acc[1]);
      acc[2] = wmma_f32(ag, *(const v2f*)(bP2 + o), acc[2]);
    }
#pragma unroll
    for (int s = 0; s < 3; ++s)
#pragma unroll
      for (int r = 0; r < 8; ++r)
        Pst[w][(r + 8 * hi) * 48 + s * 16 + nl] = acc[s][r];

    // ---- GEMM2 + skinning, two halves of the b-tile (m in [8h, 8h+8))
#pragma unroll
    for (int h = 0; h < 2; ++h) {
      v8f a2[6];
#pragma unroll
      for (int s = 0; s < 6; ++s)
#pragma unroll
        for (int r = 0; r < 8; ++r) a2[s][r] = 0.f;
#pragma unroll
      for (int ks = 0; ks < 6; ++ks) {
        int jp = 2 * ks + hi;
        const v2f* gp = G2 + jp * (SMPL_B * 12) + b0 * 12 + h * 96 + nl;
        v2f af = awf[ks];
#pragma unroll
        for (int s = 0; s < 6; ++s) a2[s] = wmma_f32(af, gp[s * 16], a2[s]);
      }
#pragma unroll
      for (int s = 0; s < 6; ++s)
#pragma unroll
        for (int r = 0; r < 8; ++r)
          Tst[w][(r + 8 * hi) * 96 + s * 16 + nl] = a2[s][r];

#pragma unroll
      for (int p = 0; p < 4; ++p) {
        int mm = 2 * p + hi;
        int m = 8 * h + mm;
        int lv = nl;
        int v = v0 + lv;
        if (v < SMPL_V) {
          int b = b0 + m;
          const v4f* tp = (const v4f*)&Tst[w][lv * 96 + mm * 12];
          v4f T0 = tp[0], T1 = tp[1], T2 = tp[2];
          float x = Pst[w][m * 48 + lv * 3 + 0] + vtl[lv * 3 + 0];
          float y = Pst[w][m * 48 + lv * 3 + 1] + vtl[lv * 3 + 1];
          float z = Pst[w][m * 48 + lv * 3 + 2] + vtl[lv * 3 + 2];
          size_t o = ((size_t)b * SMPL_V + v) * 3;
          out[o + 0] = T0.x * x + T0.y * y + T0.z * z + T0.w + trans[b * 3 + 0];
          out[o + 1] = T1.x * x + T1.y * y + T1.z * z + T1.w + trans[b * 3 + 1];
          out[o + 2] = T2.x * x + T2.y * y + T2.z * z + T2.w + trans[b * 3 + 2];
        }
      }
    }
  }
}

// ---------------------------------------------------------------- launch
extern "C" void kernel_launch(void* const* d_in, const int* in_sizes, int n_in,
                              void* d_out, int out_size, void* d_ws,
                              size_t ws_size, hipStream_t stream) {
  (void)in_sizes; (void)n_in; (void)out_size; (void)ws_size;
  const float* body_pose = (const float*)d_in[0];
  const float* betas     = (const float*)d_in[1];
  const float* trans     = (const float*)d_in[2];
  const float* v_tpl     = (const float*)d_in[3];
  const float* sdirs     = (const float*)d_in[4];
  const float* pdirs     = (const float*)d_in[5];
  const float* Jreg      = (const float*)d_in[6];
  const float* wgt       = (const float*)d_in[7];
  float* out = (float*)d_out;
  float* ws = (float*)d_ws;

  k_rots<<<dim3((SMPL_B * SMPL_J + 255) / 256), 256, 0, stream>>>(body_pose,
                                                                  betas, ws);
  k_transA<<<dim3(KK2 * SMPL_B / 256), 256, 0, stream>>>(ws);
  k_js<<<dim3((SMPL_J * 3 * 11 * 32 + 255) / 256), 256, 0, stream>>>(Jreg, sdirs,
                                                                     v_tpl, ws);
  k_chain<<<dim3(SMPL_B / 64), 64, 0, stream>>>(betas, trans, ws, out);
  k_main<<<dim3((SMPL_V + 15) / 16), 256, 0, stream>>>(ws, sdirs, pdirs, v_tpl,
                                                       wgt, trans, out);
}